// VIN_80376017977540
// MI455X (gfx1250) — compile-verified
//
#include <hip/hip_runtime.h>

typedef __attribute__((ext_vector_type(2))) float v2f;
typedef __attribute__((ext_vector_type(8))) float v8f;

typedef __attribute__((address_space(1))) int g_i32;   // global int*
typedef __attribute__((address_space(3))) int l_i32;   // LDS int*

#define IM 32
#define PW 34            // padded plane width (zero border)
#define PLANE (PW*PW)    // 1156 floats

// ---------------------------------------------------------------------------
// Precompute kernel: collapse the 150-ch hidden conv through the 1x1 readout,
// and lay out the padded 20x16 weight matrix (K = 2ch*9taps padded to 20,
// rows 10..15 / k>=18 zero) for the WMMA iteration GEMM.
// ws layout (floats): [0..8] eff_w (3x3), [9] eff_b, [16 .. 16+320) Wmat[20][16]
// ---------------------------------------------------------------------------
__global__ void vin_precompute(const float* __restrict__ h_w,
                               const float* __restrict__ h_b,
                               const float* __restrict__ r_w,
                               const float* __restrict__ q_w,
                               const float* __restrict__ w,
                               float* __restrict__ ws) {
    int t = threadIdx.x;
    if (t < 9) {
        float s = 0.f;
        for (int c = 0; c < 150; ++c) s += r_w[c] * h_w[c * 9 + t];
        ws[t] = s;
    } else if (t == 9) {
        float s = 0.f;
        for (int c = 0; c < 150; ++c) s += r_w[c] * h_b[c];
        ws[9] = s;
    }
    if (t >= 32 && t < 32 + 20 * 16) {
        int idx = t - 32;
        int k = idx >> 4, n = idx & 15;
        float val = 0.f;
        if (n < 10 && k < 18) {
            int ch = k / 9, tap = k % 9;
            val = ch ? w[n * 9 + tap] : q_w[n * 9 + tap];
        }
        ws[16 + k * 16 + n] = val;
    }
}

// ---------------------------------------------------------------------------
// Main kernel: one block (8 wave32) per image. LDS planes (zero-padded 34x34):
//   P1 = r (read-only after setup), P0/P2 = v ping-pong.
// Iteration GEMM via v_wmma_f32_16x16x4_f32 with CHANNELS ON M:
//   D[ch, px] = sum_k Wmat[ch,k] * im2col[k,px]
// so the channel-max needs only in-register VALU max + ONE xor-16 shuffle
// (vs a 32-op ds_bpermute butterfly if pixels were on M).
// ---------------------------------------------------------------------------
__global__ void __launch_bounds__(256) vin_main(const float* __restrict__ S,
                                                const float* __restrict__ fc_w,
                                                const float* __restrict__ ws,
                                                float* __restrict__ out) {
    __shared__ float P0[PLANE];   // X -> then v buffer A (starts as zeros)
    __shared__ float P1[PLANE];   // r
    __shared__ float P2[PLANE];   // v buffer B
    __shared__ float qbuf[16];

    const int img  = blockIdx.x;
    const int tid  = threadIdx.x;
    const int lane = tid & 31;
    const int wid  = tid >> 5;
    const int m    = lane & 15;   // channel (A rows) / pixel (B,D cols) index
    const int hi   = lane >> 4;   // which 16-lane half

    const float* Simg = S + (size_t)img * (IM * IM + 2);

    // ---- per-lane im2col descriptors for the 5 K-chunks (computed once) ----
    // B-fragment: lane holds im2col[k, pixel=m] for k = 4j+2*hi (+1)
    int chA[5], chB[5], dA[5], dB[5];
#pragma unroll
    for (int j = 0; j < 5; ++j) {
        int ka = 4 * j + 2 * hi;
        int kb = ka + 1;
        int k = (ka < 18) ? ka : 0;           // k>=18: A column is zero -> don't-care
        int tap = k % 9;
        chA[j] = k / 9;
        dA[j] = (tap / 3 - 1) * PW + (tap % 3 - 1);
        k = (kb < 18) ? kb : 0;
        tap = k % 9;
        chB[j] = k / 9;
        dB[j] = (tap / 3 - 1) * PW + (tap % 3 - 1);
    }
    // ---- A fragments: weights, A[ch=m, k=4j+2hi(+1)] = Wmat[k][ch] ----
    const float* Wws = ws + 16;
    v2f Af[5];
#pragma unroll
    for (int j = 0; j < 5; ++j) {
        int ka = 4 * j + 2 * hi;
        Af[j].x = Wws[ka * 16 + m];
        Af[j].y = Wws[(ka + 1) * 16 + m];
    }

    // ---- zero padded planes ----
    for (int i = tid; i < PLANE; i += 256) { P0[i] = 0.f; P1[i] = 0.f; P2[i] = 0.f; }
    __syncthreads();

    // ---- load X into P0 interior (async DMA to LDS when available) ----
#if __has_builtin(__builtin_amdgcn_global_load_async_to_lds_b32)
    for (int i = tid; i < IM * IM; i += 256) {
        int y = i >> 5, x = i & 31;
        __builtin_amdgcn_global_load_async_to_lds_b32(
            (g_i32*)(Simg + i),
            (l_i32*)&P0[(y + 1) * PW + (x + 1)],
            0, 0);
    }
#if __has_builtin(__builtin_amdgcn_s_wait_asynccnt)
    __builtin_amdgcn_s_wait_asynccnt(0);
#else
    asm volatile("s_wait_asynccnt 0x0" ::: "memory");
#endif
#else
    for (int i = tid; i < IM * IM; i += 256) {
        int y = i >> 5, x = i & 31;
        P0[(y + 1) * PW + (x + 1)] = Simg[i];
    }
#endif
    __syncthreads();

    // ---- r = conv3x3(X, eff_w) + eff_b  (150-ch layer collapsed) ----
    float ew[9];
#pragma unroll
    for (int t = 0; t < 9; ++t) ew[t] = ws[t];
    const float eb = ws[9];
    for (int i = tid; i < IM * IM; i += 256) {
        int y = i >> 5, x = i & 31;
        const float* c = &P0[(y + 1) * PW + (x + 1)];
        float s = eb;
#pragma unroll
        for (int dy = -1; dy <= 1; ++dy)
#pragma unroll
            for (int dx = -1; dx <= 1; ++dx)
                s += ew[(dy + 1) * 3 + (dx + 1)] * c[dy * PW + dx];
        P1[(y + 1) * PW + (x + 1)] = s;
    }
    __syncthreads();

    // ---- reset P0 interior: v starts at 0 (conv(0, w) == 0 == first-step semantics)
    for (int i = tid; i < IM * IM; i += 256) {
        int y = i >> 5, x = i & 31;
        P0[(y + 1) * PW + (x + 1)] = 0.f;
    }
    __syncthreads();

    // ---- 10 value-iteration steps: q = [r,v] conv W2 ; v = max_ch(q) ----
    for (int it = 0; it < 10; ++it) {
        const float* vin = (it & 1) ? P2 : P0;
        float* vout      = (it & 1) ? P0 : P2;

        // hoist per-chunk plane base pointers for this iteration
        const float* pa[5];
        const float* pb[5];
#pragma unroll
        for (int j = 0; j < 5; ++j) {
            pa[j] = (chA[j] ? vin : P1) + dA[j];
            pb[j] = (chB[j] ? vin : P1) + dB[j];
        }

        // 64 pixel-tiles of 16; 8 waves -> 8 tiles each (uniform per wave)
        for (int tt = wid; tt < 64; tt += 8) {
            const int y  = tt >> 1;
            const int x0 = (tt & 1) << 4;
            const int base = (y + 1) * PW + (x0 + 1) + m;

            v8f acc = {0.f, 0.f, 0.f, 0.f, 0.f, 0.f, 0.f, 0.f};
#pragma unroll
            for (int j = 0; j < 5; ++j) {
                v2f b;
                b.x = pa[j][base];
                b.y = pb[j][base];
                acc = __builtin_amdgcn_wmma_f32_16x16x4_f32(
                    false, Af[j], false, b, (short)0, acc, false, false);
            }

            // channel max, channels live on M:
            //   lanes 0-15 : acc[r] = q[ch=r,  px=m]  -> max of 8 in registers
            //   lanes 16-31: acc[r] = q[ch=8+r,px=m]  -> only r=0,1 valid
            float m07 = fmaxf(fmaxf(fmaxf(acc[0], acc[1]), fmaxf(acc[2], acc[3])),
                              fmaxf(fmaxf(acc[4], acc[5]), fmaxf(acc[6], acc[7])));
            float m89  = fmaxf(acc[0], acc[1]);
            float mine = hi ? m89 : m07;
            float v    = fmaxf(mine, __shfl_xor(mine, 16, 32));
            if (hi == 0)
                vout[(y + 1) * PW + (x0 + 1) + m] = v;
        }
        __syncthreads();
    }
    // after it=9 (odd), final v lives in P0
    const float* vfin = P0;

    // ---- final conv only at the gathered pixel, then FC ----
    const int s1 = (int)Simg[IM * IM];       // H index
    const int s2 = (int)Simg[IM * IM + 1];   // W index
    if (tid < 10) {
        float q = 0.f;
#pragma unroll
        for (int ch = 0; ch < 2; ++ch) {
            const float* pl = ch ? vfin : P1;
#pragma unroll
            for (int tap = 0; tap < 9; ++tap) {
                float wv = ws[16 + (ch * 9 + tap) * 16 + tid];
                q += wv * pl[(s1 + tap / 3) * PW + (s2 + tap % 3)];
            }
        }
        qbuf[tid] = q;
    }
    __syncthreads();
    if (tid < 8) {
        float s = 0.f;
#pragma unroll
        for (int n = 0; n < 10; ++n) s += fc_w[tid * 10 + n] * qbuf[n];
        out[img * 8 + tid] = s;
    }
}

// ---------------------------------------------------------------------------
extern "C" void kernel_launch(void* const* d_in, const int* in_sizes, int n_in,
                              void* d_out, int out_size, void* d_ws, size_t ws_size,
                              hipStream_t stream) {
    const float* S    = (const float*)d_in[0];
    const float* h_w  = (const float*)d_in[1];
    const float* h_b  = (const float*)d_in[2];
    const float* r_w  = (const float*)d_in[3];
    const float* q_w  = (const float*)d_in[4];
    const float* w    = (const float*)d_in[5];
    const float* fc_w = (const float*)d_in[6];
    float* ws = (float*)d_ws;

    const int Bimgs = in_sizes[0] / (IM * IM + 2);

    vin_precompute<<<1, 384, 0, stream>>>(h_w, h_b, r_w, q_w, w, ws);
    vin_main<<<Bimgs, 256, 0, stream>>>(S, fc_w, ws, (float*)d_out);
}